// ADNM_4801773437066
// MI455X (gfx1250) — compile-verified
//
#include <hip/hip_runtime.h>
#include <hip/hip_bf16.h>

#define B_   256
#define OUT_ 256
#define M_   8
#define IN_  512

typedef float v2f __attribute__((ext_vector_type(2)));
typedef float v4f __attribute__((ext_vector_type(4)));
typedef float v8f __attribute__((ext_vector_type(8)));
typedef int   v4i __attribute__((ext_vector_type(4)));

#if defined(__gfx1250__) && __has_builtin(__builtin_amdgcn_global_load_async_to_lds_b128) && \
    __has_builtin(__builtin_amdgcn_s_wait_asynccnt)
#define USE_ASYNC_LDS 1
#else
#define USE_ASYNC_LDS 0
#endif

#if USE_ASYNC_LDS
__device__ __forceinline__ void async_copy_b128(const float* g, float* l) {
    // builtin signature (from hipcc diagnostic): (int4 __device__*, int4 __shared__*, imm offset, imm cpol)
    __builtin_amdgcn_global_load_async_to_lds_b128(
        (__attribute__((address_space(1))) v4i*)(v4i*)(g),
        (__attribute__((address_space(3))) v4i*)(v4i*)(l), 0, 0);
}
#endif

__device__ __forceinline__ float fast_tanh(float x) {
#if defined(__gfx1250__) && __has_builtin(__builtin_amdgcn_tanhf)
    return __builtin_amdgcn_tanhf(x);
#elif defined(__HIP_DEVICE_COMPILE__)
    // tanh(x) = 2*sigmoid(2x)-1 ; exp(-2x) = exp2(-2*log2(e)*x)
    float e = __builtin_amdgcn_exp2f(-2.885390081777927f * x);
    return fmaf(2.0f, __builtin_amdgcn_rcpf(1.0f + e), -1.0f);
#else
    return tanhf(x);
#endif
}

__device__ __forceinline__ float fast_sigmoid(float x) {
    return fmaf(0.5f, fast_tanh(0.5f * x), 0.5f);
}

// One block per `out` column (x 2 halves of B). 8 waves/block, each wave owns
// 16 batch rows. Inner reduction over IN done with V_WMMA_F32_16X16X4_F32
// using an all-ones B matrix (row-sum on the matrix pipe). w/q slices are
// DMA'd to LDS with GLOBAL_LOAD_ASYNC_TO_LDS_B128, overlapped with the
// tanh(m) staging + block reduction.
__global__ __launch_bounds__(256) void ADNM_main_kernel(
    const float* __restrict__ x,   // [B, IN]
    const float* __restrict__ w,   // [OUT, M, IN]
    const float* __restrict__ q,   // [OUT, M, IN]
    const float* __restrict__ mm,  // [OUT, 1, IN]
    float* __restrict__ out)       // [B, OUT]
{
    __shared__ float wh_s[M_ * IN_];  // 0.5*w   (16 KB)
    __shared__ float qh_s[M_ * IN_];  // 0.5*q   (16 KB)
    __shared__ float ha_s[IN_];       // 0.5*tanh(m) (2 KB)
    __shared__ float red_s[256];      // reduction scratch (1 KB)

    const int tid = threadIdx.x;
    const int o   = blockIdx.x;  // out index

    const float* wsrc = w + (size_t)o * (M_ * IN_);
    const float* qsrc = q + (size_t)o * (M_ * IN_);

#if USE_ASYNC_LDS
    // ---- async DMA of raw w/q slices into LDS (no VGPR round-trip); this
    // proceeds in the background while we stage tanh(m) and reduce C0 ----
#pragma unroll
    for (int j = 0; j < (M_ * IN_) / (4 * 256); ++j) {  // 4 iters of b128/thread
        const int e = (tid + j * 256) * 4;              // float index, 16B aligned
        async_copy_b128(wsrc + e, &wh_s[e]);
        async_copy_b128(qsrc + e, &qh_s[e]);
    }
#else
    for (int i = tid; i < M_ * IN_; i += 256) {
        wh_s[i] = 0.5f * wsrc[i];
        qh_s[i] = 0.5f * qsrc[i];
    }
#endif

    // ---- halfA = 0.5*tanh(m); block-reduce C0 = sum(halfA) meanwhile ----
    const float* msrc = mm + (size_t)o * IN_;
    float hpart = 0.0f;
    for (int i = tid; i < IN_; i += 256) {
        float ha = 0.5f * fast_tanh(msrc[i]);
        ha_s[i] = ha;
        hpart  += ha;
    }
    red_s[tid] = hpart;
    __syncthreads();
    for (int s = 128; s > 0; s >>= 1) {
        if (tid < s) red_s[tid] += red_s[tid + s];
        __syncthreads();
    }
    const float C0 = red_s[0];  // D = C0 + sum_in halfA * tanh(u)

#if USE_ASYNC_LDS
    // async copies landed raw; halve this thread's own staged elements
    __builtin_amdgcn_s_wait_asynccnt(0);
#pragma unroll
    for (int j = 0; j < (M_ * IN_) / (4 * 256); ++j) {
        const int e = (tid + j * 256) * 4;
        v4f wv = *(v4f*)&wh_s[e];
        v4f qv = *(v4f*)&qh_s[e];
        *(v4f*)&wh_s[e] = wv * 0.5f;
        *(v4f*)&qh_s[e] = qv * 0.5f;
    }
    __syncthreads();
#endif

    // ---- per-wave tile setup ----
    const int wave  = tid >> 5;
    const int lane  = tid & 31;
    const int h     = lane >> 4;          // K-half: lanes 0-15 -> K=0,1 ; 16-31 -> K=2,3
    const int bl    = lane & 15;          // matrix row M = local batch row
    const int btile = blockIdx.y * 8 + wave;
    const int bg    = btile * 16 + bl;    // global batch row this lane feeds
    const int koff  = h * 2;

    const float* xp = x + (size_t)bg * IN_ + koff;
    __builtin_prefetch(xp, 0, 1);         // global_prefetch for the x stream

    v8f acc[M_] = {};                     // 8 WMMA accumulator tiles (one per m)
    const v2f ones = {1.0f, 1.0f};        // B matrix = all ones -> row-sum

#pragma unroll 2
    for (int c = 0; c < IN_ / 4; ++c) {
        const int ib = 4 * c + koff;
        const v2f xv = *(const v2f*)(xp + 4 * c);
        const v2f ha = *(const v2f*)(&ha_s[ib]);
#pragma unroll
        for (int m = 0; m < M_; ++m) {
            const v2f wv = *(const v2f*)(&wh_s[m * IN_ + ib]);
            const v2f qv = *(const v2f*)(&qh_s[m * IN_ + ib]);
            v2f p;
            // u = 0.5*(x*w - q) ; p = tanh(u) * halfA
            p.x = fast_tanh(fmaf(xv.x, wv.x, -qv.x)) * ha.x;
            p.y = fast_tanh(fmaf(xv.y, wv.y, -qv.y)) * ha.y;
            // D[b, *] += sum_k p  (B = ones): matrix pipe does the reduction
            acc[m] = __builtin_amdgcn_wmma_f32_16x16x4_f32(
                false, p, false, ones, (short)0, acc[m], false, false);
        }
    }

    // ---- lane L holds D rows (r + 8*h) in acc[m][r]; columns identical ----
    float ores[8];
#pragma unroll
    for (int r = 0; r < 8; ++r) {
        float s = 0.0f;
#pragma unroll
        for (int m = 0; m < M_; ++m) s += fast_sigmoid(acc[m][r] + C0);
        ores[r] = s;
    }
    if (bl == 0) {  // lanes 0 and 16 carry rows 0-7 / 8-15
        const int bb = btile * 16 + h * 8;
#pragma unroll
        for (int r = 0; r < 8; ++r) out[(size_t)(bb + r) * OUT_ + o] = ores[r];
    }
}

extern "C" void kernel_launch(void* const* d_in, const int* in_sizes, int n_in,
                              void* d_out, int out_size, void* d_ws, size_t ws_size,
                              hipStream_t stream) {
    (void)in_sizes; (void)n_in; (void)out_size; (void)d_ws; (void)ws_size;
    const float* x = (const float*)d_in[0];
    const float* w = (const float*)d_in[1];
    const float* q = (const float*)d_in[2];
    const float* m = (const float*)d_in[3];
    float* o = (float*)d_out;

    dim3 grid(OUT_, B_ / 128);  // 256 x 2 blocks
    ADNM_main_kernel<<<grid, 256, 0, stream>>>(x, w, q, m, o);
}